// PositionalEncoding_27865747817047
// MI455X (gfx1250) — compile-verified
//
#include <hip/hip_runtime.h>
#include <hip/hip_bf16.h>
#include <stdint.h>

// Problem constants from the reference (fixed for this benchmark).
#define B_DIM 16
#define S_DIM 2048
#define D_DIM 1024

typedef float f32x4 __attribute__((ext_vector_type(4)));
typedef int   i32x4 __attribute__((ext_vector_type(4)));

// ---------------------------------------------------------------------------
// Kernel 1: per-batch prefix scan of keep = (mask == 0).
// One 256-thread block (8 wave32s) per batch row. Each thread owns 8
// contiguous elements held in registers (two B128 loads), so the mask is
// read ONCE: register-count -> wave32 shfl_up scan -> wave totals through
// LDS -> per-thread exclusive base -> write ranks (two B128 stores).
// 16 blocks x 8 waves = 128 waves: ~2 memory round-trips of latency total.
// Masked positions encode as rank = -1.
// ---------------------------------------------------------------------------
__global__ void pe_ranks_kernel(const i32x4* __restrict__ mask4,
                                i32x4* __restrict__ ranks4) {
    const int b    = blockIdx.x;        // 0..15
    const int t    = threadIdx.x;       // 0..255
    const int lane = t & 31;            // wave32 lane
    const int wave = t >> 5;            // 0..7
    __shared__ int wsum[8];

    // Load this thread's 8 mask values (2 x B128), keep in registers.
    const i32x4* mp = mask4 + ((size_t)b * S_DIM) / 4 + (size_t)t * 2;
    i32x4 a = mp[0];
    i32x4 c = mp[1];
    int k[8];
    k[0] = (a.x == 0); k[1] = (a.y == 0); k[2] = (a.z == 0); k[3] = (a.w == 0);
    k[4] = (c.x == 0); k[5] = (c.y == 0); k[6] = (c.z == 0); k[7] = (c.w == 0);

    int cnt = 0;
#pragma unroll
    for (int i = 0; i < 8; ++i) cnt += k[i];

    // Inclusive wave32 scan of per-thread counts.
    int incl = cnt;
#pragma unroll
    for (int off = 1; off < 32; off <<= 1) {
        int n = __shfl_up(incl, off, 32);
        if (lane >= off) incl += n;
    }

    // Combine the 8 wave totals through LDS.
    if (lane == 31) wsum[wave] = incl;
    __syncthreads();                     // s_barrier_signal/-wait on gfx1250
    int wbase = 0;
#pragma unroll
    for (int w = 0; w < 8; ++w) wbase += (w < wave) ? wsum[w] : 0;

    // Exclusive prefix for this thread's chunk; emit ranks from registers.
    int run = wbase + incl - cnt;
    int o[8];
#pragma unroll
    for (int i = 0; i < 8; ++i) {
        run += k[i];
        o[i] = k[i] ? (run - 1) : -1;    // kept ranks are in [0, S-1]
    }

    i32x4* rp = ranks4 + ((size_t)b * S_DIM) / 4 + (size_t)t * 2;
    i32x4 r0; r0.x = o[0]; r0.y = o[1]; r0.z = o[2]; r0.w = o[3];
    i32x4 r1; r1.x = o[4]; r1.y = o[5]; r1.z = o[6]; r1.w = o[7];
    rp[0] = r0;
    rp[1] = r1;
}

// ---------------------------------------------------------------------------
// Kernel 2: out[row,:] = x[row,:] + (rank>=0 ? pe[rank,:] : 0)
// One block (256 threads = 8 waves) per (b,s) row; each thread moves one
// float4 (B128). x/out use nontemporal TH so the 256 MB stream doesn't evict
// the 16 MB L2-resident pe table; the async pe gather is tagged TH_LOAD_HT
// to bias L2 retention of pe rows. The gather uses the gfx1250 async
// global->LDS DMA path (ASYNCcnt) to overlap with the x load; each lane
// writes and reads only its own 16-byte LDS slot, so s_wait_asynccnt 0 is
// the only synchronization required (no workgroup barrier).
// ---------------------------------------------------------------------------
__global__ void pe_add_kernel(const f32x4* __restrict__ x,
                              const f32x4* __restrict__ pe,
                              const int*   __restrict__ ranks,
                              f32x4*       __restrict__ out) {
    __shared__ f32x4 tile[D_DIM / 4];            // 4 KB of the 320 KB/WGP LDS

    const int row = blockIdx.x;                  // 0 .. B*S-1
    const int t   = threadIdx.x;                 // 0 .. 255
    const size_t base = (size_t)row * (D_DIM / 4) + t;

    const int r = ranks[row];                    // block-uniform -> scalar load

    // Streaming x load: non-temporal (TH=NT), bypass long-term L2 residency.
    f32x4 xv = __builtin_nontemporal_load(x + base);

    if (r >= 0) {                                // wave-uniform branch
        // gfx1250 async data path: DMA pe row fragment into LDS (ASYNCcnt).
        const f32x4* src = pe + (size_t)r * (D_DIM / 4) + t;
        uint32_t lds_addr = (uint32_t)(uintptr_t)(&tile[t]);   // low 32 bits = LDS byte offset
        uint64_t gaddr    = (uint64_t)(uintptr_t)src;
        asm volatile("global_load_async_to_lds_b128 %0, %1, off th:TH_LOAD_HT"
                     :: "v"(lds_addr), "v"(gaddr)
                     : "memory");
        asm volatile("s_wait_asynccnt 0" ::: "memory");
        f32x4 pv = tile[t];
        xv += pv;
    }

    // Streaming store: non-temporal so output doesn't pollute L2.
    __builtin_nontemporal_store(xv, out + base);
}

// ---------------------------------------------------------------------------
// Host-side launch. Inputs (setup_inputs order): x f32[16,2048,1024],
// mask int[16,2048] (bool -> int32 per harness contract), pe f32[4096,1024].
// Output: f32[16,2048,1024]. Workspace: B*S ranks (128 KB) in d_ws.
// ---------------------------------------------------------------------------
extern "C" void kernel_launch(void* const* d_in, const int* in_sizes, int n_in,
                              void* d_out, int out_size, void* d_ws, size_t ws_size,
                              hipStream_t stream) {
    const float* x     = (const float*)d_in[0];
    const int*   mask  = (const int*)d_in[1];
    const float* pe    = (const float*)d_in[2];
    float*       out   = (float*)d_out;
    int*         ranks = (int*)d_ws;             // B*S ints = 128 KB scratch

    pe_ranks_kernel<<<dim3(B_DIM), dim3(256), 0, stream>>>(
        (const i32x4*)mask, (i32x4*)ranks);

    pe_add_kernel<<<dim3(B_DIM * S_DIM), dim3(D_DIM / 4), 0, stream>>>(
        (const f32x4*)x, (const f32x4*)pe, ranks, (f32x4*)out);
}